// StructureModule_8203387535607
// MI455X (gfx1250) — compile-verified
//
#include <hip/hip_runtime.h>
#include <hip/hip_bf16.h>
#include <math.h>

// ---------------- problem constants ----------------
#define NSEQ   768
#define C_S    384
#define C_Z    128
#define C_H    16
#define NH     12
#define PQ     4
#define PV     8
#define CO     2112          // NH*(C_Z + C_H + PV*4)
#define INF_V  100000.0f
#define EPS_V  1e-8f

// exact scale constants
#define SC_QK   0.14433756729740643f   // sqrt(1/(3*C_H))  (pre-applied to q in k_proj)
#define SC_B    0.5773502691896258f    // sqrt(1/3)
#define SC_PT   0.13608276348795434f   // sqrt(1/54)

typedef __attribute__((ext_vector_type(2))) float v2f;
typedef __attribute__((ext_vector_type(8))) float v8f;

__device__ __forceinline__ v8f wmma4(v2f a, v2f b, v8f c) {
  // V_WMMA_F32_16X16X4_F32 : D = A(16x4) * B(4x16) + C(16x16)
  return __builtin_amdgcn_wmma_f32_16x16x4_f32(false, a, false, b, (short)0, c, false, false);
}

__device__ __forceinline__ v8f v8zero() {
  v8f z = {0.f,0.f,0.f,0.f,0.f,0.f,0.f,0.f};
  return z;
}

// ---------------- workspace layout (floats) ----------------
#define WS_Q     0          // [N][192]   (pre-scaled by SC_QK)
#define WS_K     147456     // [N][192]
#define WS_V     294912     // [N][192]
#define WS_QP    442368     // [N][144]
#define WS_KP    552960     // [N][144]
#define WS_VP    663552     // [N][NH][32]  padded, cols 24..31 zero
#define WS_CAT   958464     // [N][2112]
#define WS_LOG   2580480    // [NH][N][N]
#define WS_ZROW  9658368    // [N] zeros (dead-lane A rows for k_out_pair)
// total 9,659,136 floats = 38.6 MB

// ============================================================
// K1: fused input projections + rigid transform of points
// ============================================================
__global__ __launch_bounds__(256) void k_proj(
    const float* __restrict__ s, const float* __restrict__ R, const float* __restrict__ t,
    const float* __restrict__ Wq,  const float* __restrict__ bq,
    const float* __restrict__ Wkv, const float* __restrict__ bkv,
    const float* __restrict__ Wqp, const float* __restrict__ bqp,
    const float* __restrict__ Wkvp,const float* __restrict__ bkvp,
    float* __restrict__ q, float* __restrict__ k, float* __restrict__ v,
    float* __restrict__ qpts, float* __restrict__ kpts, float* __restrict__ vpts,
    float* __restrict__ zrow0)
{
  const int n = blockIdx.x;
  __shared__ float srow[C_S];
  __shared__ float proj[1152];   // 192 q | 384 kv | 144 qp | 432 kvp

  if (threadIdx.x == 0) zrow0[n] = 0.f;          // build the zero row once
  for (int i = threadIdx.x; i < C_S; i += 256) srow[i] = s[n*C_S + i];
  // zero-fill the padded tail of v_pts rows (cols 24..31 for each head)
  for (int u = threadIdx.x; u < NH*8; u += 256)
    vpts[n*384 + (u >> 3)*32 + 24 + (u & 7)] = 0.f;
  __syncthreads();

  for (int u = threadIdx.x; u < 1152; u += 256) {
    const float* W; int col, width; float bias;
    if (u < 192)      { W = Wq;   col = u;       width = 192; bias = bq[col];   }
    else if (u < 576) { W = Wkv;  col = u - 192; width = 384; bias = bkv[col];  }
    else if (u < 720) { W = Wqp;  col = u - 576; width = 144; bias = bqp[col];  }
    else              { W = Wkvp; col = u - 720; width = 432; bias = bkvp[col]; }
    float acc = bias;
    for (int c = 0; c < C_S; ++c) acc += srow[c] * W[c*width + col];
    proj[u] = acc;
  }
  __syncthreads();

  // scalar heads: q (pre-scaled) / k / v
  for (int u = threadIdx.x; u < 576; u += 256) {
    if (u < 192) {
      q[n*192 + u] = proj[u] * SC_QK;
    } else {
      int u2 = u - 192, h = u2 >> 5, w = u2 & 31;
      if (w < 16) k[n*192 + h*16 + w]        = proj[u];
      else        v[n*192 + h*16 + (w - 16)] = proj[u];
    }
  }

  // point heads: rigid apply  x' = R x + t
  const int tg = threadIdx.x;
  if (tg < 192) {
    float R0[9], tt[3];
    #pragma unroll
    for (int i = 0; i < 9; ++i) R0[i] = R[n*9 + i];
    tt[0] = t[n*3]; tt[1] = t[n*3+1]; tt[2] = t[n*3+2];
    if (tg < 48) {                       // q_pts: 48 points
      int j = tg;
      float px = proj[576 + j], py = proj[576 + 48 + j], pz = proj[576 + 96 + j];
      float ox = R0[0]*px + R0[1]*py + R0[2]*pz + tt[0];
      float oy = R0[3]*px + R0[4]*py + R0[5]*pz + tt[1];
      float oz = R0[6]*px + R0[7]*py + R0[8]*pz + tt[2];
      float* d = qpts + n*144 + j*3;     // j = h*PQ + p
      d[0] = ox; d[1] = oy; d[2] = oz;
    } else {                             // kv_pts: 144 points
      int j = tg - 48;
      float px = proj[720 + j], py = proj[720 + 144 + j], pz = proj[720 + 288 + j];
      float ox = R0[0]*px + R0[1]*py + R0[2]*pz + tt[0];
      float oy = R0[3]*px + R0[4]*py + R0[5]*pz + tt[1];
      float oz = R0[6]*px + R0[7]*py + R0[8]*pz + tt[2];
      int h = j / 12, r = j % 12;
      float* d = (r < 4) ? (kpts + n*144 + (h*4 + r)*3)
                         : (vpts + n*384 + h*32 + (r - 4)*3);   // padded layout
      d[0] = ox; d[1] = oy; d[2] = oz;
    }
  }
}

// ============================================================
// K2: attention logits for one 16x16 (q,k) tile, all 12 heads
//   phase 1: b = z@Wb via WMMA -> LDS, Wb fragments preloaded to
//            registers (branchless z-stream: b64 load + wmma only)
//   phase 2: QK^T via WMMA into acc[12]
//   phase 3: fold b from LDS
//   phase 4: point-attention + mask, h-outer to bound live values
// ============================================================
__global__ __launch_bounds__(32, 4) void k_logits(
    const float* __restrict__ q, const float* __restrict__ k,
    const float* __restrict__ qpts, const float* __restrict__ kpts,
    const float* __restrict__ z, const float* __restrict__ Wb, const float* __restrict__ bb,
    const float* __restrict__ head_weights, const float* __restrict__ mask,
    float* __restrict__ logits)
{
  const int q0 = blockIdx.x * 16, k0 = blockIdx.y * 16;
  const int lane = threadIdx.x, l16 = lane & 15, half = lane >> 4;

  __shared__ float bb_lds[16][16][16];          // [i][j][h] 16 KB

  // ---- preload all Wb B-fragments (invariant over pt): 32 chunks x v2f
  v2f WbF[32];
  #pragma unroll
  for (int cc = 0; cc < 32; ++cc) {
    const int c = cc*4 + half*2;
    WbF[cc].x = (l16 < NH) ? Wb[(size_t)c      * NH + l16] : 0.f;
    WbF[cc].y = (l16 < NH) ? Wb[(size_t)(c + 1)* NH + l16] : 0.f;
  }

  // ---- phase 1: pair bias b = z @ Wb  (M=16 k-cols of one q-row, N=heads, K=128)
  #pragma unroll 1
  for (int pt = 0; pt < 16; ++pt) {             // pt = q-row within tile
    v8f bacc[4];
    #pragma unroll
    for (int u = 0; u < 4; ++u) bacc[u] = v8zero();
    const float* zrow = z + ((size_t)(q0 + pt) * NSEQ + (k0 + l16)) * C_Z;
    if (pt < 15) {  // prefetch next q-row of the z tile (stream from HBM)
      __builtin_prefetch(zrow + (size_t)NSEQ * C_Z + half * 64, 0, 1);
    }
    #pragma unroll
    for (int cc = 0; cc < 32; ++cc) {           // branchless: b64 load + wmma
      const int c = cc*4 + half*2;
      v2f A; A.x = zrow[c]; A.y = zrow[c + 1];
      bacc[cc & 3] = wmma4(A, WbF[cc], bacc[cc & 3]);
    }
    v8f bs = (bacc[0] + bacc[1]) + (bacc[2] + bacc[3]);
    #pragma unroll
    for (int m = 0; m < 8; ++m) bb_lds[pt][m + half*8][l16] = bs[m];
  }
  __syncthreads();

  // ---- phase 2: QK^T via WMMA (q pre-scaled; K = C_H = 16 -> 4 chunks)
  v8f acc[NH];
  #pragma unroll
  for (int h = 0; h < NH; ++h) acc[h] = v8zero();
  const float* qrow = q + (size_t)(q0 + l16) * 192;
  const float* krow = k + (size_t)(k0 + l16) * 192;
  #pragma unroll
  for (int h = 0; h < NH; ++h) {
    #pragma unroll
    for (int kc = 0; kc < 4; ++kc) {
      const int c = kc*4 + half*2;
      v2f A; A.x = qrow[h*16 + c]; A.y = qrow[h*16 + c + 1];
      v2f B; B.x = krow[h*16 + c]; B.y = krow[h*16 + c + 1];
      acc[h] = wmma4(A, B, acc[h]);
    }
  }

  // ---- phase 3: fold b (element (i=pt, j=l16) lives in half==pt>>3, vgpr pt&7)
  #pragma unroll
  for (int pt = 0; pt < 16; ++pt) {
    if (half == (pt >> 3)) {
      const int m = pt & 7;
      #pragma unroll
      for (int h = 0; h < NH; ++h)
        acc[h][m] += SC_B * (bb_lds[pt][l16][h] + bb[h]);
    }
  }

  // ---- phase 4: point attention + mask + store (h-outer bounds live regs)
  const int j = l16;
  const float mj = mask[k0 + j];
  const float* kpb = kpts + (size_t)(k0 + j) * 144;
  #pragma unroll
  for (int h = 0; h < NH; ++h) {
    const float x = head_weights[h];
    const float hwh = ((x > 20.f) ? x : log1pf(expf(x))) * SC_PT;
    float kx[PQ][3];
    #pragma unroll
    for (int p = 0; p < PQ; ++p) {
      kx[p][0] = kpb[(h*PQ + p)*3];
      kx[p][1] = kpb[(h*PQ + p)*3 + 1];
      kx[p][2] = kpb[(h*PQ + p)*3 + 2];
    }
    #pragma unroll
    for (int m = 0; m < 8; ++m) {
      const int i = m + half*8;
      const float* qp = qpts + (size_t)(q0 + i) * 144 + h*PQ*3;
      float sd = 0.f;
      #pragma unroll
      for (int p = 0; p < PQ; ++p) {
        float dx = qp[p*3]     - kx[p][0];
        float dy = qp[p*3 + 1] - kx[p][1];
        float dz = qp[p*3 + 2] - kx[p][2];
        sd += dx*dx + dy*dy + dz*dz;
      }
      const float mi = mask[q0 + i];
      logits[(size_t)h*NSEQ*NSEQ + (size_t)(q0 + i)*NSEQ + (k0 + j)] =
          acc[h][m] + INF_V * (mi * mj - 1.0f) - 0.5f * hwh * sd;
    }
  }
}

// ============================================================
// K3: row softmax over k (rows = NH*N, length N=768)
// ============================================================
__global__ __launch_bounds__(256) void k_softmax(float* __restrict__ logits)
{
  float* row = logits + (size_t)blockIdx.x * NSEQ;
  __shared__ float red[256];
  const int tx = threadIdx.x;
  float x0 = row[tx], x1 = row[tx + 256], x2 = row[tx + 512];
  red[tx] = fmaxf(x0, fmaxf(x1, x2));
  __syncthreads();
  for (int s = 128; s > 0; s >>= 1) {
    if (tx < s) red[tx] = fmaxf(red[tx], red[tx + s]);
    __syncthreads();
  }
  const float mx = red[0];
  __syncthreads();
  x0 = expf(x0 - mx); x1 = expf(x1 - mx); x2 = expf(x2 - mx);
  red[tx] = x0 + x1 + x2;
  __syncthreads();
  for (int s = 128; s > 0; s >>= 1) {
    if (tx < s) red[tx] += red[tx + s];
    __syncthreads();
  }
  const float inv = 1.0f / red[0];
  row[tx] = x0 * inv; row[tx + 256] = x1 * inv; row[tx + 512] = x2 * inv;
}

// ============================================================
// K4a: o = a@v, o_pt = R^T(a@v_pts - t), norms -> cat
//   one wave per (q-tile, head); 6 independent WMMA chains; padded
//   v_pts layout makes all B loads branchless
// ============================================================
__global__ __launch_bounds__(32) void k_out_ov(
    const float* __restrict__ probs, const float* __restrict__ v, const float* __restrict__ vpts,
    const float* __restrict__ R, const float* __restrict__ t, float* __restrict__ cat)
{
  const int q0 = blockIdx.x * 16, h = blockIdx.y;
  const int lane = threadIdx.x, l16 = lane & 15, half = lane >> 4;

  v8f ao[2], ap1[2], ap2[2];
  #pragma unroll
  for (int u = 0; u < 2; ++u) { ao[u] = v8zero(); ap1[u] = v8zero(); ap2[u] = v8zero(); }
  const float* arow = probs + (size_t)h*NSEQ*NSEQ + (size_t)(q0 + l16)*NSEQ;

  #pragma unroll 1
  for (int k0 = 0; k0 < NSEQ; k0 += 8) {
    #pragma unroll
    for (int u = 0; u < 2; ++u) {
      const int kr = k0 + u*4 + half*2;
      v2f A; A.x = arow[kr]; A.y = arow[kr + 1];
      const float* v0 = v    + (size_t)kr*192 + h*16;
      const float* p0 = vpts + (size_t)kr*384 + h*32;     // padded row
      v2f B0; B0.x = v0[l16];      B0.y = v0[192 + l16];
      v2f B1; B1.x = p0[l16];      B1.y = p0[384 + l16];
      v2f B2; B2.x = p0[16 + l16]; B2.y = p0[384 + 16 + l16];
      ao[u]  = wmma4(A, B0, ao[u]);
      ap1[u] = wmma4(A, B1, ap1[u]);
      ap2[u] = wmma4(A, B2, ap2[u]);
    }
  }
  const v8f aoS  = ao[0]  + ao[1];
  const v8f ap1S = ap1[0] + ap1[1];
  const v8f ap2S = ap2[0] + ap2[1];

  __shared__ float opt[16 * 24];
  #pragma unroll
  for (int m = 0; m < 8; ++m) {
    const int i = m + half*8;
    cat[(size_t)(q0 + i)*CO + h*16 + l16] = aoS[m];       // o block
    opt[i*24 + l16] = ap1S[m];
    if (l16 < 8) opt[i*24 + 16 + l16] = ap2S[m];
  }
  __syncthreads();

  // inverse rigid + norm: 16 rows * 8 points
  #pragma unroll
  for (int tl = 0; tl < 4; ++tl) {
    const int idx = lane + tl*32;          // 0..127
    const int i = idx >> 3, p = idx & 7;
    const int n = q0 + i;
    const float gx = opt[i*24 + p*3]     - t[n*3];
    const float gy = opt[i*24 + p*3 + 1] - t[n*3 + 1];
    const float gz = opt[i*24 + p*3 + 2] - t[n*3 + 2];
    const float* Rn = R + n*9;
    const float ux = Rn[0]*gx + Rn[3]*gy + Rn[6]*gz;      // R^T
    const float uy = Rn[1]*gx + Rn[4]*gy + Rn[7]*gz;
    const float uz = Rn[2]*gx + Rn[5]*gy + Rn[8]*gz;
    const float nrm = sqrtf(ux*ux + uy*uy + uz*uz + EPS_V);
    const size_t base = (size_t)n * CO;
    const int hp = h*PV + p;
    cat[base + 192 + hp] = ux;
    cat[base + 288 + hp] = uy;
    cat[base + 384 + hp] = uz;
    cat[base + 480 + hp] = nrm;
  }
}

// ============================================================
// K4b: o_pair[q,h,cz] = sum_k a[h,q,k] * z[q,k,cz]
//   one wave per (q, 16-col tile); 4 chains; dead A-lanes read a
//   zero row so the z stream is branchless
// ============================================================
__global__ __launch_bounds__(32) void k_out_pair(
    const float* __restrict__ probs, const float* __restrict__ z,
    const float* __restrict__ zrow0, float* __restrict__ cat)
{
  const int qn = blockIdx.x, cz0 = blockIdx.y * 16;
  const int lane = threadIdx.x, l16 = lane & 15, half = lane >> 4;

  v8f acc[4];
  #pragma unroll
  for (int u = 0; u < 4; ++u) acc[u] = v8zero();
  const float* arow = (l16 < NH)
      ? (probs + (size_t)l16*NSEQ*NSEQ + (size_t)qn*NSEQ)
      : zrow0;                                   // guaranteed zeros
  const float* zq = z + (size_t)qn * NSEQ * C_Z;

  #pragma unroll 1
  for (int k0 = 0; k0 < NSEQ; k0 += 16) {
    if (k0 + 64 < NSEQ) {  // prefetch the z stream ~4 iterations ahead
      __builtin_prefetch(&zq[(size_t)(k0 + 64 + lane) * C_Z + cz0], 0, 1);
    }
    #pragma unroll
    for (int u = 0; u < 4; ++u) {
      const int kr = k0 + u*4 + half*2;
      v2f A; A.x = arow[kr]; A.y = arow[kr + 1];
      v2f B;
      B.x = zq[(size_t)kr      * C_Z + cz0 + l16];
      B.y = zq[(size_t)(kr + 1)* C_Z + cz0 + l16];
      acc[u] = wmma4(A, B, acc[u]);
    }
  }
  const v8f aS = (acc[0] + acc[1]) + (acc[2] + acc[3]);
  #pragma unroll
  for (int m = 0; m < 8; ++m) {
    const int hh = m + half*8;
    if (hh < NH)
      cat[(size_t)qn*CO + 576 + hh*C_Z + cz0 + l16] = aS[m];
  }
}

// ============================================================
// K5: out = cat @ Wout + bout   (768x2112 @ 2112x384), 4 chains
// ============================================================
__global__ __launch_bounds__(32) void k_final(
    const float* __restrict__ cat, const float* __restrict__ Wout,
    const float* __restrict__ bout, float* __restrict__ out)
{
  const int q0 = blockIdx.x * 16, c0 = blockIdx.y * 16;
  const int lane = threadIdx.x, l16 = lane & 15, half = lane >> 4;

  v8f acc[4];
  #pragma unroll
  for (int u = 0; u < 4; ++u) acc[u] = v8zero();
  const float* arow = cat + (size_t)(q0 + l16) * CO;

  #pragma unroll 1
  for (int k0 = 0; k0 < CO; k0 += 16) {
    #pragma unroll
    for (int u = 0; u < 4; ++u) {
      const int kr = k0 + u*4 + half*2;
      v2f A; A.x = arow[kr]; A.y = arow[kr + 1];
      v2f B;
      B.x = Wout[(size_t)kr      * C_S + c0 + l16];
      B.y = Wout[(size_t)(kr + 1)* C_S + c0 + l16];
      acc[u] = wmma4(A, B, acc[u]);
    }
  }
  const v8f aS = (acc[0] + acc[1]) + (acc[2] + acc[3]);
  #pragma unroll
  for (int m = 0; m < 8; ++m) {
    const int i = m + half*8;
    out[(size_t)(q0 + i)*C_S + c0 + l16] = aS[m] + bout[c0 + l16];
  }
}

// ============================================================
extern "C" void kernel_launch(void* const* d_in, const int* in_sizes, int n_in,
                              void* d_out, int out_size, void* d_ws, size_t ws_size,
                              hipStream_t stream) {
  const float* s    = (const float*)d_in[0];
  const float* z    = (const float*)d_in[1];
  const float* R    = (const float*)d_in[2];
  const float* t    = (const float*)d_in[3];
  const float* mask = (const float*)d_in[4];
  const float* Wq   = (const float*)d_in[5];
  const float* bq   = (const float*)d_in[6];
  const float* Wkv  = (const float*)d_in[7];
  const float* bkv  = (const float*)d_in[8];
  const float* Wqp  = (const float*)d_in[9];
  const float* bqp  = (const float*)d_in[10];
  const float* Wkvp = (const float*)d_in[11];
  const float* bkvp = (const float*)d_in[12];
  const float* Wb   = (const float*)d_in[13];
  const float* bb   = (const float*)d_in[14];
  const float* hwts = (const float*)d_in[15];
  const float* Wout = (const float*)d_in[16];
  const float* bout = (const float*)d_in[17];

  float* ws   = (float*)d_ws;
  float* q    = ws + WS_Q;
  float* k    = ws + WS_K;
  float* v    = ws + WS_V;
  float* qpts = ws + WS_QP;
  float* kpts = ws + WS_KP;
  float* vpts = ws + WS_VP;
  float* cat  = ws + WS_CAT;
  float* logits = ws + WS_LOG;
  float* zrow0  = ws + WS_ZROW;

  k_proj<<<NSEQ, 256, 0, stream>>>(s, R, t, Wq, bq, Wkv, bkv, Wqp, bqp, Wkvp, bkvp,
                                   q, k, v, qpts, kpts, vpts, zrow0);
  k_logits<<<dim3(NSEQ/16, NSEQ/16), 32, 0, stream>>>(q, k, qpts, kpts, z, Wb, bb,
                                                      hwts, mask, logits);
  k_softmax<<<NH * NSEQ, 256, 0, stream>>>(logits);
  k_out_ov<<<dim3(NSEQ/16, NH), 32, 0, stream>>>(logits, v, vpts, R, t, cat);
  k_out_pair<<<dim3(NSEQ, C_Z/16), 32, 0, stream>>>(logits, z, zrow0, cat);
  k_final<<<dim3(NSEQ/16, C_S/16), 32, 0, stream>>>(cat, Wout, bout, (float*)d_out);
}